// ULTIMUS_21861383537086
// MI455X (gfx1250) — compile-verified
//
#include <hip/hip_runtime.h>
#include <math.h>

// out = softmax-attention "ULTIMUS": q,k,v = x@W{q,k,v}^T ; scores = q^T k /sqrt(8)
// attn = softmax(scores, axis=1); out = (v@attn)@Wo^T
// Restructured:  A2 = attn@Wo^T (8x48);  out = v @ A2   (v cached in ws)
// Fallback (small ws): M = Wv^T@attn@Wo^T (48x48); out = x @ M

typedef float v2f __attribute__((ext_vector_type(2)));
typedef float v8f __attribute__((ext_vector_type(8)));

#define NB1 1024          // blocks in pass-1 / partial-score slots
#define XS_STRIDE 52      // padded LDS row stride (floats) -> conflict-free A-frag reads
#define QK_STRIDE 20      // padded LDS row stride for Q|K tile

// LDS layout (floats), one buffer so xs provably sits at a high offset:
//   [0      .. 2560)  qk   : 8 waves x 16*QK_STRIDE
//   [2560   .. 3072)  wsum : 8 waves x 64
//   [3072   .. 9728)  xs   : 8 waves x 16*XS_STRIDE
#define SMEM_FLOATS 9728

__device__ __forceinline__ v8f wmma4(v2f a, v2f b, v8f c) {
    // D(16x16,f32) = A(16x4,f32) x B(4x16,f32) + C
    return __builtin_amdgcn_wmma_f32_16x16x4_f32(false, a, false, b, (short)0, c,
                                                 false, false);
}

// Async-stage a 16x48 f32 tile (3072B) into this wave's padded LDS region.
// INST_OFFSET is added to BOTH global and LDS addresses, so each LDS address
// register is pre-compensated by -512*i; one global base pointer serves all 6.
__device__ __forceinline__ void async_stage6(const unsigned* l, const float* gp) {
    asm volatile(
        "global_load_async_to_lds_b128 %0, %6, off\n\t"
        "global_load_async_to_lds_b128 %1, %6, off offset:512\n\t"
        "global_load_async_to_lds_b128 %2, %6, off offset:1024\n\t"
        "global_load_async_to_lds_b128 %3, %6, off offset:1536\n\t"
        "global_load_async_to_lds_b128 %4, %6, off offset:2048\n\t"
        "global_load_async_to_lds_b128 %5, %6, off offset:2560"
        :
        : "v"(l[0]), "v"(l[1]), "v"(l[2]), "v"(l[3]), "v"(l[4]), "v"(l[5]),
          "v"(gp)
        : "memory");
}

// ---------------- Pass 1: q,k,v + score partials ----------------
__global__ void __launch_bounds__(256)
k1_qkv_scores(const float* __restrict__ x, const float* __restrict__ Wq,
              const float* __restrict__ Wk, const float* __restrict__ Wv,
              float* __restrict__ vout, float* __restrict__ partial,
              int ntiles, int store_v)
{
    __shared__ __align__(16) float smem[SMEM_FLOATS];

    const int tid  = threadIdx.x;
    const int lane = tid & 31;
    const int widx = tid >> 5;
    const int lo   = lane & 15;
    const int hi   = lane >> 4;   // 0|1
    const int hi2  = hi << 1;

    float* qk   = smem + widx * (16 * QK_STRIDE);
    float* wsum = smem + 2560;
    float* xsw  = smem + 3072 + widx * (16 * XS_STRIDE);

    // Loop-invariant pre-compensated LDS dest registers for the async stage.
    unsigned ldsreg[6];
#pragma unroll
    for (int i = 0; i < 6; ++i) {
        int f4  = i * 32 + lane;             // 0..191 float4 chunks
        int row = f4 / 12;
        int col = (f4 % 12) * 4;
        ldsreg[i] = (unsigned)(uintptr_t)(xsw + row * XS_STRIDE + col)
                    - (unsigned)(512 * i);
    }

    // B fragments: B1 = [Wq;Wk]^T chunks (16 useful cols), B2 = [Wv;0]^T chunks.
    // B(k',n): n = lane%16, k' = (lane/16)*2 + j   (j = vgpr 0/1)
    v2f b1[12], b2[12];
#pragma unroll
    for (int kc = 0; kc < 12; ++kc) {
        const float* wrow = (lo < 8) ? (Wq + lo * 48) : (Wk + (lo - 8) * 48);
        b1[kc].x = wrow[kc * 4 + hi2 + 0];
        b1[kc].y = wrow[kc * 4 + hi2 + 1];
        if (lo < 8) {
            b2[kc].x = Wv[lo * 48 + kc * 4 + hi2 + 0];
            b2[kc].y = Wv[lo * 48 + kc * 4 + hi2 + 1];
        } else {
            b2[kc].x = 0.f; b2[kc].y = 0.f;
        }
    }

    // each lane accumulates 2 elements of the wave's 8x8 score matrix
    const int aidx = lane >> 2;         // 0..7 (q column)
    const int b0   = (lane & 3) << 1;   // 0,2,4,6 (k column pair)
    float sc0 = 0.f, sc1 = 0.f;

    const int gwave  = blockIdx.x * 8 + widx;
    const int nwaves = gridDim.x * 8;

    if (gwave < ntiles)  // prologue fetch, tile 0 of this wave
        async_stage6(ldsreg, x + (size_t)gwave * 768 + lane * 4);

    for (int tile = gwave; tile < ntiles; tile += nwaves) {
        const int next = tile + nwaves;

        asm volatile("s_wait_asynccnt 0x0" ::: "memory");  // tile data in LDS
        __builtin_amdgcn_wave_barrier();

        v8f c1 = {0, 0, 0, 0, 0, 0, 0, 0};   // Q(0..7)|K(8..15)
        v8f c2 = {0, 0, 0, 0, 0, 0, 0, 0};   // V(0..7)|zero
#pragma unroll
        for (int kc = 0; kc < 12; ++kc) {
            // A(m,k'): lane = m + 16*(k'>=2), vgpr = k'&1
            v2f a = *(const v2f*)&xsw[lo * XS_STRIDE + kc * 4 + hi2];
            c1 = wmma4(a, b1[kc], c1);
            c2 = wmma4(a, b2[kc], c2);
        }

        // xs DS reads retired -> safe to let the async engine overwrite it;
        // prefetch next tile, overlapped with the v-store + score phase below.
        asm volatile("s_wait_dscnt 0x0" ::: "memory");
        if (next < ntiles)
            async_stage6(ldsreg, x + (size_t)next * 768 + lane * 4);

        // store v rows (C layout: vgpr r -> row r + 8*hi, col = lo)
        if (store_v && lo < 8) {
            float* vp = vout + ((size_t)tile * 16 + hi * 8) * 8 + lo;
#pragma unroll
            for (int r = 0; r < 8; ++r) vp[r * 8] = c2[r];
        }

        // Q|K tile to LDS for cross-lane outer-product accumulation
#pragma unroll
        for (int r = 0; r < 8; ++r)
            qk[(hi * 8 + r) * QK_STRIDE + lo] = c1[r];
        __builtin_amdgcn_wave_barrier();

#pragma unroll
        for (int r = 0; r < 16; ++r) {
            float qa = qk[r * QK_STRIDE + aidx];
            v2f  kb = *(const v2f*)&qk[r * QK_STRIDE + 8 + b0];
            sc0 = fmaf(qa, kb.x, sc0);
            sc1 = fmaf(qa, kb.y, sc1);
        }
        __builtin_amdgcn_wave_barrier();
    }

    wsum[widx * 64 + aidx * 8 + b0]     = sc0;
    wsum[widx * 64 + aidx * 8 + b0 + 1] = sc1;
    __syncthreads();
    if (tid < 64) {   // deterministic fixed-order cross-wave reduce
        float s = 0.f;
#pragma unroll
        for (int w = 0; w < 8; ++w) s += wsum[w * 64 + tid];
        partial[(size_t)blockIdx.x * 64 + tid] = s;
    }
}

// ---------------- Pass 2: reduce + softmax + A2 / M ----------------
__global__ void __launch_bounds__(256)
k2_softmax_mats(const float* __restrict__ partial, const float* __restrict__ Wo,
                const float* __restrict__ Wv, float* __restrict__ A2g,
                float* __restrict__ Mg)
{
    __shared__ float red[4][64];
    __shared__ float sc[64];
    __shared__ float attn[64];
    __shared__ float a2s[8 * 48];
    const int t = threadIdx.x;
    const int e = t & 63, ch = t >> 6;

    float s = 0.f;
    for (int i = ch; i < NB1; i += 4) s += partial[(size_t)i * 64 + e];
    red[ch][e] = s;
    __syncthreads();
    if (t < 64)
        sc[t] = (red[0][t] + red[1][t] + red[2][t] + red[3][t]) *
                0.35355339059327373f;  // 1/sqrt(8)
    __syncthreads();
    if (t < 8) {  // softmax over axis=1 (columns j) per row t
        float m = sc[t * 8];
        for (int j = 1; j < 8; ++j) m = fmaxf(m, sc[t * 8 + j]);
        float e8[8], sum = 0.f;
        for (int j = 0; j < 8; ++j) { e8[j] = expf(sc[t * 8 + j] - m); sum += e8[j]; }
        float inv = 1.0f / sum;
        for (int j = 0; j < 8; ++j) attn[t * 8 + j] = e8[j] * inv;
    }
    __syncthreads();
    // A2[a][d] = sum_b attn[a][b] * Wo[d][b]
    for (int idx = t; idx < 8 * 48; idx += 256) {
        int a = idx / 48, d = idx % 48;
        float v = 0.f;
        for (int b = 0; b < 8; ++b) v = fmaf(attn[a * 8 + b], Wo[d * 8 + b], v);
        a2s[idx] = v;
        A2g[idx] = v;
    }
    __syncthreads();
    // fallback M[c][d] = sum_a Wv[a][c] * A2[a][d]
    for (int idx = t; idx < 48 * 48; idx += 256) {
        int c = idx / 48, d = idx % 48;
        float v = 0.f;
        for (int a = 0; a < 8; ++a) v = fmaf(Wv[a * 48 + c], a2s[a * 48 + d], v);
        Mg[idx] = v;
    }
}

// ---------------- Pass 3: out[N,48] = S[N,KS] @ B[KS,48] ----------------
template <int KS>
__global__ void __launch_bounds__(256)
k3_gemm(const float* __restrict__ S, const float* __restrict__ B,
        float* __restrict__ out, int ntiles)
{
    const int lane = threadIdx.x & 31;
    const int widx = threadIdx.x >> 5;
    const int lo   = lane & 15;
    const int hi   = lane >> 4;
    const int hi2  = hi << 1;
    constexpr int KC = KS / 4;

    v2f bf[KC][3];  // whole B matrix resident in VGPRs
#pragma unroll
    for (int kc = 0; kc < KC; ++kc)
#pragma unroll
        for (int nt = 0; nt < 3; ++nt) {
            bf[kc][nt].x = B[(kc * 4 + hi2 + 0) * 48 + nt * 16 + lo];
            bf[kc][nt].y = B[(kc * 4 + hi2 + 1) * 48 + nt * 16 + lo];
        }

    const int gwave  = blockIdx.x * 8 + widx;
    const int nwaves = gridDim.x * 8;

    for (int tile = gwave; tile < ntiles; tile += nwaves) {
        const float* st = S + (size_t)tile * 16 * KS + lo * KS + hi2;
        v8f c0 = {0, 0, 0, 0, 0, 0, 0, 0}, c1 = c0, c2 = c0;
#pragma unroll
        for (int kc = 0; kc < KC; ++kc) {
            v2f a = *(const v2f*)(st + kc * 4);
            c0 = wmma4(a, bf[kc][0], c0);
            c1 = wmma4(a, bf[kc][1], c1);
            c2 = wmma4(a, bf[kc][2], c2);
        }
        float* ot = out + ((size_t)tile * 16 + hi * 8) * 48 + lo;
#pragma unroll
        for (int r = 0; r < 8; ++r) {
            ot[r * 48 +  0] = c0[r];
            ot[r * 48 + 16] = c1[r];
            ot[r * 48 + 32] = c2[r];
        }
    }
}

extern "C" void kernel_launch(void* const* d_in, const int* in_sizes, int n_in,
                              void* d_out, int out_size, void* d_ws, size_t ws_size,
                              hipStream_t stream)
{
    (void)n_in; (void)out_size;
    // setup_inputs order: x, Wk, Wq, Wv, Wo
    const float* x  = (const float*)d_in[0];
    const float* Wk = (const float*)d_in[1];
    const float* Wq = (const float*)d_in[2];
    const float* Wv = (const float*)d_in[3];
    const float* Wo = (const float*)d_in[4];
    const int N      = in_sizes[0] / 48;
    const int ntiles = N / 16;   // N = 1<<20 -> 65536 tiles

    float* ws      = (float*)d_ws;
    float* partial = ws;                        // NB1*64 floats
    float* A2g     = ws + NB1 * 64;             // 384 floats
    float* Mg      = A2g + 512;                 // 2304 floats
    float* vbuf    = ws + NB1 * 64 + 4096;      // N*8 floats (if it fits)
    const size_t need_v =
        ((size_t)(NB1 * 64 + 4096) + (size_t)N * 8) * sizeof(float);
    const int use_v = (ws_size >= need_v) ? 1 : 0;

    k1_qkv_scores<<<NB1, 256, 0, stream>>>(x, Wq, Wk, Wv, vbuf, partial, ntiles,
                                           use_v);
    k2_softmax_mats<<<1, 256, 0, stream>>>(partial, Wo, Wv, A2g, Mg);
    if (use_v)
        k3_gemm<8><<<1024, 256, 0, stream>>>(vbuf, A2g, (float*)d_out, ntiles);
    else
        k3_gemm<48><<<1024, 256, 0, stream>>>(x, Mg, (float*)d_out, ntiles);
}